// CachedAttention_28355374088878
// MI455X (gfx1250) — compile-verified
//
#include <hip/hip_runtime.h>
#include <hip/hip_bf16.h>
#include <math.h>

typedef float v2f __attribute__((ext_vector_type(2)));
typedef float v4f __attribute__((ext_vector_type(4)));
typedef float v8f __attribute__((ext_vector_type(8)));

#define B_ 16
#define D_ 1024
#define N_ 16
#define H_ 64
#define S_ 4096
#define NH_ 1024            // N_*H_
#define SPLIT_ 8
#define SCHUNK_ (S_ / SPLIT_)   // 512 S-rows per block

// ---------------------------------------------------------------------------
// Kernel 1: QKV projections via V_WMMA_F32_16X16X4_F32.
// X [16,1024] x W [1024,1024] -> [16,1024] (+bias).  One wave per 16x16 tile:
// 3 matrices * 64 column tiles = 192 blocks of 32 threads.
// A-layout (16x4 f32): lane holds A[lane&15][(lane>>4)*2 + j] in reg j.
// B-layout (4x16 f32): lane holds B[(lane>>4)*2 + j][lane&15] in reg j.
// C/D (16x16 f32):     reg v, lane -> row v + 8*(lane>>4), col lane&15.
// ---------------------------------------------------------------------------
__global__ __launch_bounds__(32)
void qkv_proj_wmma(const float* __restrict__ from_t,
                   const float* __restrict__ to_t,
                   const float* __restrict__ wq, const float* __restrict__ bq,
                   const float* __restrict__ wk, const float* __restrict__ bk,
                   const float* __restrict__ wv, const float* __restrict__ bv,
                   float* __restrict__ q_ws, float* __restrict__ k_ws,
                   float* __restrict__ v_ws) {
    const int mat   = blockIdx.x >> 6;          // 0=q, 1=k, 2=v
    const int nbase = (blockIdx.x & 63) << 4;   // output column tile
    const float* X    = (mat == 0) ? from_t : to_t;
    const float* W    = (mat == 0) ? wq : (mat == 1) ? wk : wv;
    const float* bias = (mat == 0) ? bq : (mat == 1) ? bk : bv;
    float* outp       = (mat == 0) ? q_ws : (mat == 1) ? k_ws : v_ws;

    const int lane = threadIdx.x;
    const int r    = lane & 15;
    const int half = lane >> 4;

    v8f acc = {};
    const float* xrow = X + r * D_;          // row = batch index
    const float* wcol = W + nbase + r;       // W row stride = NH_ = 1024
    #pragma unroll 4
    for (int k = 0; k < D_; k += 4) {
        const int ka = k + half * 2;
        v2f a, b;
        a.x = xrow[ka];
        a.y = xrow[ka + 1];
        b.x = wcol[(long)ka * NH_];
        b.y = wcol[(long)(ka + 1) * NH_];
        acc = __builtin_amdgcn_wmma_f32_16x16x4_f32(false, a, false, b,
                                                    (short)0, acc, false, false);
    }
    const float bb = bias[nbase + r];
    #pragma unroll
    for (int v = 0; v < 8; ++v) {
        const int m = v + 8 * half;          // batch row
        outp[m * NH_ + nbase + r] = acc[v] + bb;
    }
}

// ---------------------------------------------------------------------------
// Kernel 2: fused cache-update + flash-attention streaming pass.
// Grid = B*N*SPLIT = 2048 blocks, 256 threads (8 wave32).
// Each block owns one (b, n, s-chunk of 512): streams key/value rows once,
// writes key_t/value_t (non-temporal b128), computes online softmax partials.
// Thread layout: 16 row-groups of 16 lanes; each lane owns a 4-float h-quad.
// ---------------------------------------------------------------------------
__global__ __launch_bounds__(256)
void cache_attn_pass(const float* __restrict__ cache_key,
                     const float* __restrict__ cache_value,
                     const int*   __restrict__ mask,      // [B,1,S]
                     const int*   __restrict__ step_p,
                     const float* __restrict__ q_ws,
                     const float* __restrict__ k_ws,
                     const float* __restrict__ v_ws,
                     float* __restrict__ key_out,
                     float* __restrict__ val_out,
                     float* __restrict__ part_ml,         // [2048][2]
                     float* __restrict__ part_acc)        // [2048][64]
{
    __shared__ float lds_m[16];
    __shared__ float lds_l[16];
    __shared__ v4f   lds_acc[16][16];

    const int bid    = blockIdx.x;
    const int split  = bid & (SPLIT_ - 1);
    const int n      = (bid >> 3) & 15;
    const int b      = bid >> 7;
    const int t      = threadIdx.x;
    const int g      = t >> 4;          // row-group: 16 s-rows in flight
    const int lane16 = t & 15;          // h-quad owner
    const int step   = *step_p;

    const long hoff = (long)b * NH_ + n * H_ + lane16 * 4;
    const v4f qq = *(const v4f*)(q_ws + hoff);
    const v4f kn = *(const v4f*)(k_ws + hoff);
    const v4f vn = *(const v4f*)(v_ws + hoff);

    float m = -INFINITY, l = 0.0f;
    v4f acc = {};

    const int s0 = split * SCHUNK_;
    for (int it = 0; it < SCHUNK_ / 16; ++it) {
        const int  s   = s0 + it * 16 + g;
        const long idx = (((long)b * S_ + s) * N_ + n) * H_ + lane16 * 4;

        v4f kq = __builtin_nontemporal_load((const v4f*)(cache_key   + idx));
        v4f vq = __builtin_nontemporal_load((const v4f*)(cache_value + idx));
        if (s == step) { kq += kn; vq += vn; }     // one-hot cache scatter
        __builtin_nontemporal_store(kq, (v4f*)(key_out + idx));
        __builtin_nontemporal_store(vq, (v4f*)(val_out + idx));

        // dot(key_row, q) across the 16-lane group (butterfly stays in half-wave)
        float p = kq.x * qq.x + kq.y * qq.y + kq.z * qq.z + kq.w * qq.w;
        p += __shfl_xor(p, 1, 16);
        p += __shfl_xor(p, 2, 16);
        p += __shfl_xor(p, 4, 16);
        p += __shfl_xor(p, 8, 16);

        const float score = p * 0.125f +
                            (1.0f - (float)mask[b * S_ + s]) * (-10000.0f);

        // online softmax update
        const float m_new = fmaxf(m, score);
        const float resc  = __expf(m - m_new);
        const float pr    = __expf(score - m_new);
        l   = l * resc + pr;
        acc = acc * resc + pr * vq;
        m   = m_new;
    }

    // merge the 16 row-groups of this block
    if (lane16 == 0) { lds_m[g] = m; lds_l[g] = l; }
    lds_acc[g][lane16] = acc;
    __syncthreads();

    if (t < 64) {
        float mt = lds_m[0];
        #pragma unroll
        for (int i = 1; i < 16; ++i) mt = fmaxf(mt, lds_m[i]);
        float lt = 0.0f, a = 0.0f;
        #pragma unroll
        for (int i = 0; i < 16; ++i) {
            const float w = __expf(lds_m[i] - mt);
            lt += lds_l[i] * w;
            a  += ((const float*)lds_acc[i])[t] * w;
        }
        part_acc[(long)bid * 64 + t] = a;
        if (t == 0) { part_ml[bid * 2] = mt; part_ml[bid * 2 + 1] = lt; }
    }
}

// ---------------------------------------------------------------------------
// Kernel 3: merge the SPLIT_ partials per (b,n) and emit out[b,0,n,h].
// ---------------------------------------------------------------------------
__global__ __launch_bounds__(64)
void attn_combine(const float* __restrict__ part_ml,
                  const float* __restrict__ part_acc,
                  float* __restrict__ out)   // [B,1,N,H]
{
    const int bn   = blockIdx.x;      // b*16 + n
    const int t    = threadIdx.x;     // h
    const int base = bn * SPLIT_;

    float mt = -INFINITY;
    #pragma unroll
    for (int i = 0; i < SPLIT_; ++i) mt = fmaxf(mt, part_ml[(base + i) * 2]);
    float lt = 0.0f, a = 0.0f;
    #pragma unroll
    for (int i = 0; i < SPLIT_; ++i) {
        const float w = __expf(part_ml[(base + i) * 2] - mt);
        lt += part_ml[(base + i) * 2 + 1] * w;
        a  += part_acc[(long)(base + i) * 64 + t] * w;
    }
    out[bn * H_ + t] = a / lt;
}

// ---------------------------------------------------------------------------
extern "C" void kernel_launch(void* const* d_in, const int* in_sizes, int n_in,
                              void* d_out, int out_size, void* d_ws, size_t ws_size,
                              hipStream_t stream) {
    const float* from_t  = (const float*)d_in[0];
    const float* to_t    = (const float*)d_in[1];
    const int*   mask    = (const int*)  d_in[2];
    const float* cache_k = (const float*)d_in[3];
    const float* cache_v = (const float*)d_in[4];
    const float* wq      = (const float*)d_in[5];
    const float* bq      = (const float*)d_in[6];
    const float* wk      = (const float*)d_in[7];
    const float* bk      = (const float*)d_in[8];
    const float* wv      = (const float*)d_in[9];
    const float* bv      = (const float*)d_in[10];
    const int*   step    = (const int*)  d_in[11];

    // outputs, concatenated flat in reference return order
    float* out     = (float*)d_out;                        // [16,1,16,64]
    float* key_out = out + (long)B_ * N_ * H_;             // [16,4096,16,64]
    float* val_out = key_out + (long)B_ * S_ * N_ * H_;

    // workspace carve-up (floats)
    float* ws       = (float*)d_ws;
    float* q_ws     = ws;                    // 16384
    float* k_ws     = q_ws + B_ * NH_;       // 16384
    float* v_ws     = k_ws + B_ * NH_;       // 16384
    float* part_ml  = v_ws + B_ * NH_;       // 2048 * 2
    float* part_acc = part_ml + (B_ * N_ * SPLIT_) * 2;    // 2048 * 64

    qkv_proj_wmma<<<3 * (NH_ / 16), 32, 0, stream>>>(
        from_t, to_t, wq, bq, wk, bk, wv, bv, q_ws, k_ws, v_ws);

    cache_attn_pass<<<B_ * N_ * SPLIT_, 256, 0, stream>>>(
        cache_k, cache_v, mask, step, q_ws, k_ws, v_ws,
        key_out, val_out, part_ml, part_acc);

    attn_combine<<<B_ * N_, 64, 0, stream>>>(part_ml, part_acc, out);
}